// LocalGatedPropagation_22222160789649
// MI455X (gfx1250) — compile-verified
//
#include <hip/hip_runtime.h>
#include <math.h>

// ---------------------------------------------------------------------------
// LocalGatedPropagation for MI455X (gfx1250, wave32).
// Dense contractions -> V_WMMA_F32_16X16X4_F32 (fp32, exact).
// Windowed correlation / softmax / aggregation / dwconv -> VALU.
// GEMM waves compute a 64x32 tile (4x2 WMMA subtiles): clamped indices,
// branchless inner loop, running pointers, explicit global-AS loads
// (global_load_b64/b32, LOADcnt-only waits; no flat/DScnt coupling).
// ---------------------------------------------------------------------------

#define HWp   900      // 30*30
#define NBATCH 2
#define NHEAD 8
#define WSQ   225      // 15*15
#define DATTc 32
#define INV_SQRT32 0.17677669529663687f

#define MT 4           // 16-row subtiles per wave  (64 rows)
#define NT 2           // 16-col subtiles per wave  (32 cols)

typedef __attribute__((ext_vector_type(2))) float v2f;
typedef __attribute__((ext_vector_type(8))) float v8f;

// Explicit global-address-space pointer types (force global_load/global_store).
typedef const float __attribute__((address_space(1)))* gcfp;
typedef       float __attribute__((address_space(1)))* gfp;
typedef const v2f   __attribute__((address_space(1)))* gcv2p;

// ---------------------------------------------------------------------------
// Generic one-wave 64x32-tile GEMM:  D[m,p] = sum_k A[m,k]*B[k,p] + bias[m]
//   A row-major (M x K), lda = K.  B channel-major (K x P), ldb = P.
//   mode 0: store D[m*P+p]        (per-z offset Dzs)
//   mode 1: SiLU epilogue, same store
//   mode 2: final projection store D[(p*NBATCH+z)*M + m]  (D is d_out base)
// Per-z operand offsets:  A += (z%Azm)*Azs,  B += z*Bzs,  bias += (z%bzm)*bzs
// Out-of-range rows/cols are CLAMPED on load (garbage only reaches D
// entries that the guarded epilogue never stores) -> no exec masking.
// ---------------------------------------------------------------------------
__global__ __launch_bounds__(32) void gemm_wmma_64x32(
    const float* __restrict__ Abase, const float* __restrict__ Bbase,
    const float* __restrict__ biasBase, float* __restrict__ Dbase,
    int M, int K, int P,
    long Azs, int Azm, long Bzs, long Dzs, int bzs, int bzm, int mode)
{
    const int z = blockIdx.z;
    gcfp A    = (gcfp)(Abase    + (long)(z % Azm) * Azs);
    gcfp B    = (gcfp)(Bbase    + (long)z * Bzs);
    gcfp bias = (gcfp)(biasBase + (long)(z % bzm) * bzs);
    gfp  D    = (gfp) (Dbase    + (long)z * Dzs);

    const int lane = threadIdx.x & 31;
    const int hi   = lane >> 4;       // lane group 0/1
    const int lo   = lane & 15;
    const int tM   = blockIdx.x * (16 * MT);
    const int tP   = blockIdx.y * (16 * NT);

    // Clamped per-lane source row/col (ISA 7.12.2 fp32 A 16x4 / B 4x16 layout:
    // lanes 0-15 hold K=0,1; lanes 16-31 hold K=2,3).
    gcfp Ap[MT];
    #pragma unroll
    for (int i = 0; i < MT; ++i) {
        int am = tM + 16 * i + lo;
        am = (am < M) ? am : (M - 1);
        Ap[i] = A + (long)am * K + hi * 2;
    }
    gcfp Bp[NT];
    #pragma unroll
    for (int j = 0; j < NT; ++j) {
        int bp = tP + 16 * j + lo;
        bp = (bp < P) ? bp : (P - 1);
        Bp[j] = B + bp + (long)(hi * 2) * P;
    }

    v8f acc[MT][NT];
    #pragma unroll
    for (int i = 0; i < MT; ++i)
        #pragma unroll
        for (int j = 0; j < NT; ++j)
            acc[i][j] = (v8f){0.f,0.f,0.f,0.f,0.f,0.f,0.f,0.f};

    #pragma unroll 2
    for (int k0 = 0; k0 < K; k0 += 4) {
        v2f a[MT], b[NT];
        #pragma unroll
        for (int i = 0; i < MT; ++i) {
            a[i] = *(gcv2p)Ap[i];           // global_load_b64 (8B aligned)
            Ap[i] += 4;
        }
        #pragma unroll
        for (int j = 0; j < NT; ++j) {
            b[j].x = Bp[j][0];
            b[j].y = Bp[j][(long)P];
            Bp[j] += (long)4 * P;
        }
        #pragma unroll
        for (int i = 0; i < MT; ++i)
            #pragma unroll
            for (int j = 0; j < NT; ++j)
                acc[i][j] = __builtin_amdgcn_wmma_f32_16x16x4_f32(
                    false, a[i], false, b[j], (short)0, acc[i][j], false, false);
    }

    // C/D layout: VGPR r -> M = r + 8*hi, N = lo.
    #pragma unroll
    for (int j = 0; j < NT; ++j) {
        const int p = tP + 16 * j + lo;
        if (p < P) {
            #pragma unroll
            for (int i = 0; i < MT; ++i) {
                #pragma unroll
                for (int r = 0; r < 8; ++r) {
                    const int m = tM + 16 * i + hi * 8 + r;
                    if (m < M) {
                        float val = acc[i][j][r] + bias[m];
                        if (mode == 1) val = val / (1.f + __expf(-val));   // SiLU
                        if (mode == 2) D[((long)p * NBATCH + z) * M + m] = val;
                        else           D[(long)m * P + p] = val;
                    }
                }
            }
        }
    }
}

// ---------------------------------------------------------------------------
// Windowed correlation + boundary mask.  logits already holds rel-pos term.
//   logits[z,o,p] += valid ? (q[.,p] . q_feat[., p+off]) / sqrt(32) : -1e8
// z = n*8+h, one thread per (z,o,p).
// ---------------------------------------------------------------------------
__global__ void corr_mask_kernel(const float* __restrict__ qkfeat,
                                 float* __restrict__ logits)
{
    const long total = (long)NBATCH * NHEAD * WSQ * HWp;
    const long idx = (long)blockIdx.x * blockDim.x + threadIdx.x;
    if (idx >= total) return;

    const int p = (int)(idx % HWp);
    long t = idx / HWp;
    const int o = (int)(t % WSQ);
    const int z = (int)(t / WSQ);
    const int n = z >> 3, h = z & 7;

    const int y = p / 30, x = p % 30;
    const int ky = y + (o / 15) - 7;
    const int kx = x + (o % 15) - 7;

    float add;
    if ((unsigned)ky < 30u && (unsigned)kx < 30u) {
        const int kp = ky * 30 + kx;
        gcfp qb = (gcfp)(qkfeat + (long)n * 230400 + (long)h * DATTc * HWp);
        float s = 0.f;
        #pragma unroll
        for (int c = 0; c < DATTc; ++c)
            s += qb[c * HWp + p] * qb[c * HWp + kp];
        add = s * INV_SQRT32;
    } else {
        add = -1e8f;
    }
    logits[idx] += add;
}

// ---------------------------------------------------------------------------
// Softmax over the 225 window positions: one wave per (z,p). 16*900 waves.
// ---------------------------------------------------------------------------
__global__ __launch_bounds__(256) void softmax_kernel(
    const float* __restrict__ logits, float* __restrict__ attn)
{
    const int wave = (int)((blockIdx.x * blockDim.x + threadIdx.x) >> 5);
    const int lane = threadIdx.x & 31;
    if (wave >= NBATCH * NHEAD * HWp) return;

    const int p = wave % HWp;
    const int z = wave / HWp;
    gcfp src = (gcfp)(logits + (long)z * (WSQ * HWp) + p);

    float vals[8];
    float mx = -INFINITY;
    #pragma unroll
    for (int i = 0; i < 8; ++i) {
        const int o = lane + i * 32;
        vals[i] = (o < WSQ) ? src[(long)o * HWp] : -INFINITY;
        mx = fmaxf(mx, vals[i]);
    }
    for (int off = 16; off; off >>= 1) mx = fmaxf(mx, __shfl_xor(mx, off, 32));

    float sum = 0.f;
    #pragma unroll
    for (int i = 0; i < 8; ++i) {
        const int o = lane + i * 32;
        vals[i] = (o < WSQ) ? __expf(vals[i] - mx) : 0.f;
        sum += vals[i];
    }
    for (int off = 16; off; off >>= 1) sum += __shfl_xor(sum, off, 32);
    const float inv = 1.f / sum;

    gfp dst = (gfp)(attn + (long)z * (WSQ * HWp) + p);
    #pragma unroll
    for (int i = 0; i < 8; ++i) {
        const int o = lane + i * 32;
        if (o < WSQ) dst[(long)o * HWp] = vals[i] * inv;
    }
}

// ---------------------------------------------------------------------------
// Local aggregation (attn x unfolded v) + head-interleave + gating with uu.
// vv/uu raw channel for (head h, interleaved ch c): (c>>4)*128 + h*16 + (c&15).
// Output gate[n, h*32+c, p] channel-major for the depthwise conv.
// ---------------------------------------------------------------------------
__global__ void agg_gate_kernel(const float* __restrict__ attn,
                                const float* __restrict__ vvraw,
                                const float* __restrict__ uuraw,
                                float* __restrict__ gate)
{
    const int total = NBATCH * NHEAD * DATTc * HWp;   // 460800
    const int idx = blockIdx.x * blockDim.x + threadIdx.x;
    if (idx >= total) return;

    const int p = idx % HWp;
    int t = idx / HWp;
    const int c = t % 32;  t /= 32;
    const int h = t % 8;
    const int n = t / 8;

    const int rawc = ((c >> 4) * 128) + h * 16 + (c & 15);
    gcfp vb = (gcfp)(vvraw + (long)n * 230400 + (long)rawc * HWp);
    gcfp ab = (gcfp)(attn + (long)(n * NHEAD + h) * (WSQ * HWp) + p);

    const int y = p / 30, x = p % 30;
    float s = 0.f;
    for (int o = 0; o < WSQ; ++o) {
        const int ky = y + (o / 15) - 7;
        const int kx = x + (o % 15) - 7;
        if ((unsigned)ky < 30u && (unsigned)kx < 30u)
            s += ab[(long)o * HWp] * vb[ky * 30 + kx];
    }
    const float g = uuraw[(long)n * 230400 + (long)rawc * HWp + p];
    const int ch = h * 32 + c;
    gate[(long)n * 230400 + (long)ch * HWp + p] = s * g;
}

// ---------------------------------------------------------------------------
// Depthwise 5x5 conv, pad 2, no bias.
// ---------------------------------------------------------------------------
__global__ void dwconv_kernel(const float* __restrict__ gate,
                              const float* __restrict__ Wdw,
                              float* __restrict__ dwout)
{
    const int total = NBATCH * 256 * HWp;
    const int idx = blockIdx.x * blockDim.x + threadIdx.x;
    if (idx >= total) return;

    const int p  = idx % HWp;
    const int ch = (idx / HWp) % 256;
    const int n  = idx / (HWp * 256);
    const int y = p / 30, x = p % 30;

    gcfp src = (gcfp)(gate + (long)n * 230400 + (long)ch * HWp);
    gcfp w   = (gcfp)(Wdw + ch * 25);
    float s = 0.f;
    #pragma unroll
    for (int ky = 0; ky < 5; ++ky) {
        const int iy = y + ky - 2;
        if ((unsigned)iy >= 30u) continue;
        #pragma unroll
        for (int kx = 0; kx < 5; ++kx) {
            const int ix = x + kx - 2;
            if ((unsigned)ix < 30u)
                s += src[iy * 30 + ix] * w[ky * 5 + kx];
        }
    }
    dwout[idx] = s;
}

// ---------------------------------------------------------------------------
extern "C" void kernel_launch(void* const* d_in, const int* in_sizes, int n_in,
                              void* d_out, int out_size, void* d_ws, size_t ws_size,
                              hipStream_t stream) {
    (void)in_sizes; (void)n_in; (void)out_size; (void)ws_size;

    const float* q     = (const float*)d_in[0];
    // d_in[1] (k) is unused: reference overwrites it with linear_QK(q).
    const float* v     = (const float*)d_in[2];
    const float* u     = (const float*)d_in[3];
    const float* Wqk   = (const float*)d_in[4];
    const float* bqk   = (const float*)d_in[5];
    const float* Wv    = (const float*)d_in[6];
    const float* bv    = (const float*)d_in[7];
    const float* Wu    = (const float*)d_in[8];
    const float* bu    = (const float*)d_in[9];
    const float* Wrel  = (const float*)d_in[10];
    const float* brel  = (const float*)d_in[11];
    const float* Wdw   = (const float*)d_in[12];
    const float* Wproj = (const float*)d_in[13];
    const float* bproj = (const float*)d_in[14];

    float* out  = (float*)d_out;          // (hw, n, 128) = 230400
    float* attn = out + 230400;           // (n, 8, 225, 900) = 3240000

    float* ws     = (float*)d_ws;
    float* qkfeat = ws;                   // 460800 (reused as gate later)
    float* vvraw  = ws + 460800;          // 460800 (reused as dwout later)
    float* uuraw  = ws + 921600;          // 460800
    float* logits = ws + 1382400;         // 3240000
    float* gate   = qkfeat;               // alias: qkfeat dead after corr
    float* dwout  = vvraw;                // alias: vvraw dead after agg_gate

    const dim3 wave(32, 1, 1);
    const int PT = (HWp + 16 * NT - 1) / (16 * NT);   // 29 col-tiles

    // 1) qk_feat[n] = Wqk . q[n] + bqk          (M=256,K=256,P=900, Z=n)
    gemm_wmma_64x32<<<dim3(4, PT, 2), wave, 0, stream>>>(
        Wqk, q, bqk, qkfeat, 256, 256, 900,
        0L, 1, 230400L, 230400L, 0, 1, 0);

    // 2) vv = SiLU(grouped 1x1)                 (M=128,K=128, Z=n*2+g)
    gemm_wmma_64x32<<<dim3(2, PT, 4), wave, 0, stream>>>(
        Wv, v, bv, vvraw, 128, 128, 900,
        16384L, 2, 115200L, 115200L, 128, 2, 1);

    // 3) uu = SiLU(grouped 1x1)
    gemm_wmma_64x32<<<dim3(2, PT, 4), wave, 0, stream>>>(
        Wu, u, bu, uuraw, 128, 128, 900,
        16384L, 2, 115200L, 115200L, 128, 2, 1);

    // 4) rel-pos term -> logits                 (M=225,K=32, Z=n*8+h)
    gemm_wmma_64x32<<<dim3(4, PT, 16), wave, 0, stream>>>(
        Wrel, qkfeat, brel, logits, 225, 32, 900,
        7200L, 8, 28800L, 202500L, 225, 8, 0);

    // 5) windowed q.k correlation + boundary mask (accumulates into logits)
    corr_mask_kernel<<<(3240000 + 255) / 256, 256, 0, stream>>>(qkfeat, logits);

    // 6) softmax over 225 -> local_attn (second output region of d_out)
    softmax_kernel<<<1800, 256, 0, stream>>>(logits, attn);

    // 7) aggregation + gating -> gate (channel-major)
    agg_gate_kernel<<<1800, 256, 0, stream>>>(attn, vvraw, uuraw, gate);

    // 8) depthwise 5x5
    dwconv_kernel<<<1800, 256, 0, stream>>>(gate, Wdw, dwout);

    // 9) final projection, stored as (hw, n, 128)  (M=128,K=256, Z=n, mode 2)
    gemm_wmma_64x32<<<dim3(2, PT, 2), wave, 0, stream>>>(
        Wproj, dwout, bproj, out, 128, 256, 900,
        0L, 1, 230400L, 0L, 0, 1, 2);
}